// KVQuantizer_2525440770925
// MI455X (gfx1250) — compile-verified
//
#include <hip/hip_runtime.h>
#include <stdint.h>

// KV-cache diff quantizer for MI455X (gfx1250, wave32).
// GROUP_SIZE == D == 128 -> every quant group is a contiguous 512B vector
// feat[h,s,:]; every (head,chunk) tile is 8KB contiguous. Bandwidth-bound:
// ~268MB traffic -> ~11.5us floor @ 23.3 TB/s. CDNA5 path:
// global_load_async_to_lds_b128 (ASYNCcnt) bulk tile staging + NT stores.

#define H_HEADS 32
#define DDIM    128
#define CHUNK   16
#define KEEP    64      // keep top-64 |deq| per 128-group (prune_budget=0.5)
#define EPSQ    1e-5f

typedef float f32x4 __attribute__((ext_vector_type(4)));

__device__ __forceinline__ void store_nt4(float* p, float a, float b, float c, float d) {
  f32x4 v = {a, b, c, d};
  __builtin_nontemporal_store(v, (f32x4*)p);
}

__device__ __forceinline__ float wave_max(float v) {
#pragma unroll
  for (int d = 16; d > 0; d >>= 1) v = fmaxf(v, __shfl_xor(v, d, 32));
  return v;
}
__device__ __forceinline__ float wave_min(float v) {
#pragma unroll
  for (int d = 16; d > 0; d >>= 1) v = fminf(v, __shfl_xor(v, d, 32));
  return v;
}
// extract 8-bit bin p (0..15) from 4 packed words, p may be runtime
__device__ __forceinline__ int byte_of(unsigned w0, unsigned w1, unsigned w2,
                                       unsigned w3, int p) {
  unsigned w = (p < 8) ? ((p < 4) ? w0 : w1) : ((p < 12) ? w2 : w3);
  return (int)((w >> ((p & 3) * 8)) & 0xFFu);
}

__global__ __launch_bounds__(512) void kvq_kernel(const float* __restrict__ feat,
                                                  const int* __restrict__ dlen_p,
                                                  float* __restrict__ out,
                                                  int S) {
  __shared__ float tile[CHUNK * DDIM];  // 8 KB: one (chunk,head) tile
  __shared__ float basebuf[DDIM];       // dequantized base row

  const int t    = threadIdx.x;   // 0..511
  const int lane = t & 31;
  const int wv   = t >> 5;        // token index within chunk, 0..15
  const int c = blockIdx.x;       // chunk index (no div needed: 2D grid)
  const int h = blockIdx.y;       // head index
  const size_t g0 = ((size_t)h * S + (size_t)c * CHUNK) * DDIM;

  // ---- CDNA5 async bulk copy: 8KB contiguous tile, one b128 per thread ----
  {
    unsigned ldsoff = (unsigned)(uintptr_t)(&tile[t * 4]);
    unsigned long long ga = (unsigned long long)(feat + g0 + (size_t)t * 4);
    asm volatile("global_load_async_to_lds_b128 %0, %1, off"
                 :: "v"(ldsoff), "v"(ga) : "memory");
    asm volatile("s_wait_asynccnt 0" ::: "memory");
  }
  __syncthreads();  // all waves' async writes now visible

  const int diff_len = *dlen_p;
  const bool quant_chunk = (c * CHUNK) < diff_len;  // uniform per block

  // my row (4 consecutive floats per lane)
  float v[4];
#pragma unroll
  for (int j = 0; j < 4; ++j) v[j] = tile[wv * DDIM + lane * 4 + j];

  if (!quant_chunk) {  // passthrough region (s >= diff_len)
    store_nt4(out + g0 + (size_t)wv * DDIM + lane * 4, v[0], v[1], v[2], v[3]);
    return;
  }

  // ---- wave 0: 8-bit asymmetric quant/dequant of chunk-base row ----
  float basev[4];
  if (wv == 0) {
    float mx = fmaxf(fmaxf(v[0], v[1]), fmaxf(v[2], v[3]));
    float mn = fminf(fminf(v[0], v[1]), fminf(v[2], v[3]));
    mx = wave_max(mx); mn = wave_min(mn);
    float s8  = fmaxf((mx - mn) * (1.0f / 255.0f), EPSQ);
    float rs8 = __builtin_amdgcn_rcpf(s8);   // one rcp per group
#pragma unroll
    for (int j = 0; j < 4; ++j) {
      float qf = fminf(fmaxf(rintf((v[j] - mn) * rs8), 0.0f), 255.0f);
      basev[j] = qf * s8 + mn;
      basebuf[lane * 4 + j] = basev[j];
    }
  }
  __syncthreads();

  if (wv == 0) {  // output row 0 = base (diff row 0 quantizes to exactly 0)
    store_nt4(out + g0 + lane * 4, basev[0], basev[1], basev[2], basev[3]);
    return;
  }

  // ---- diff rows: 4-bit asymmetric quant + top-64-|deq| prune ----
  float b[4], dfv[4];
#pragma unroll
  for (int j = 0; j < 4; ++j) { b[j] = basebuf[lane * 4 + j]; dfv[j] = v[j] - b[j]; }
  float mx = fmaxf(fmaxf(dfv[0], dfv[1]), fmaxf(dfv[2], dfv[3]));
  float mn = fminf(fminf(dfv[0], dfv[1]), fminf(dfv[2], dfv[3]));
  mx = wave_max(mx); mn = wave_min(mn);
  const float s4  = fmaxf((mx - mn) * (1.0f / 15.0f), EPSQ);
  const float rs4 = __builtin_amdgcn_rcpf(s4);

  int q[4]; float dq[4];
  unsigned hw0 = 0, hw1 = 0, hw2 = 0, hw3 = 0;  // 16 bins x 8 bits
#pragma unroll
  for (int j = 0; j < 4; ++j) {
    float qf = fminf(fmaxf(rintf((dfv[j] - mn) * rs4), 0.0f), 15.0f);
    q[j] = (int)qf;
    dq[j] = qf * s4 + mn;
    unsigned bump = 1u << ((q[j] & 3) * 8);
    switch (q[j] >> 2) {
      case 0: hw0 += bump; break; case 1: hw1 += bump; break;
      case 2: hw2 += bump; break; default: hw3 += bump; break;
    }
  }
  // inclusive lane-scan of packed histogram (for tie-break by index)
  unsigned i0 = hw0, i1 = hw1, i2 = hw2, i3 = hw3;
#pragma unroll
  for (int d = 1; d < 32; d <<= 1) {
    unsigned u0 = __shfl_up(i0, d, 32), u1 = __shfl_up(i1, d, 32);
    unsigned u2 = __shfl_up(i2, d, 32), u3 = __shfl_up(i3, d, 32);
    if (lane >= d) { i0 += u0; i1 += u1; i2 += u2; i3 += u3; }
  }
  const unsigned t0 = __shfl(i0, 31, 32), t1 = __shfl(i1, 31, 32);
  const unsigned t2 = __shfl(i2, 31, 32), t3 = __shfl(i3, 31, 32);
  const unsigned e0 = i0 - hw0, e1 = i1 - hw1, e2 = i2 - hw2, e3 = i3 - hw3;

  // per-level "strictly greater count": level p handled by lane (lane&15);
  // |deq| takes <=16 distinct values per group, so this is a 16x16 compare.
  int gr;
  {
    const int p = lane & 15;
    const float kp = fabsf((float)p * s4 + mn);
    gr = 0;
#pragma unroll
    for (int r = 0; r < 16; ++r) {
      float kr = fabsf((float)r * s4 + mn);
      bool g = (r != p) && ((kr > kp) || (kr == kp && r > p));
      if (g) gr += byte_of(t0, t1, t2, t3, r);
    }
  }

  float outv[4];
#pragma unroll
  for (int j = 0; j < 4; ++j) {
    const int qq = q[j];
    int greater = __shfl(gr, qq, 32);          // dynamic-lane gather (ds_bpermute)
    int eq_tot    = byte_of(t0, t1, t2, t3, qq);
    int eq_before = byte_of(e0, e1, e2, e3, qq);
#pragma unroll
    for (int j2 = 0; j2 < 4; ++j2)
      if (j2 < j && q[j2] == qq) eq_before++;
    int eq_after = eq_tot - 1 - eq_before;     // ties: higher index survives
    bool keep = (greater + eq_after) < KEEP;
    outv[j] = b[j] + (keep ? dq[j] : 0.0f);
  }
  store_nt4(out + g0 + (size_t)wv * DDIM + lane * 4, outv[0], outv[1], outv[2], outv[3]);
}

extern "C" void kernel_launch(void* const* d_in, const int* in_sizes, int n_in,
                              void* d_out, int out_size, void* d_ws, size_t ws_size,
                              hipStream_t stream) {
  const float* feat = (const float*)d_in[0];
  const int*   dlen = (const int*)d_in[1];
  float*       out  = (float*)d_out;
  const int S = in_sizes[0] / (H_HEADS * DDIM);      // 8192
  dim3 grid(S / CHUNK, H_HEADS);                     // 512 x 32 blocks, no div in kernel
  kvq_kernel<<<grid, 512, 0, stream>>>(feat, dlen, out, S);
}